// MedianBlur_8237747273973
// MI455X (gfx1250) — compile-verified
//
#include <hip/hip_runtime.h>
#include <cstdint>

#define IMG 512
#define TX  64          // output tile width
#define TY  8           // output tile height
#define HX  (TX + 4)    // 68 halo width  (floats)
#define HY  (TY + 4)    // 12 halo height
#define NCHUNK ((HX * HY) / 2)   // 408 b64 chunks

#define AS1 __attribute__((address_space(1)))
#define AS3 __attribute__((address_space(3)))

typedef int   v2i __attribute__((vector_size(8)));    // native 2 x i32
typedef float v2f __attribute__((ext_vector_type(2))); // native 2 x f32

// ---- gfx1250 async global->LDS staging ------------------------------------
__device__ __forceinline__ void async_copy_b64(const float* gptr, float* lptr) {
#if defined(__AMDGCN__) && __has_builtin(__builtin_amdgcn_global_load_async_to_lds_b64)
    __builtin_amdgcn_global_load_async_to_lds_b64(
        (AS1 v2i*)gptr, (AS3 v2i*)lptr, /*offset=*/0, /*cpol=*/0);
#elif defined(__AMDGCN__) && __has_builtin(__builtin_amdgcn_global_load_async_to_lds_b32)
    __builtin_amdgcn_global_load_async_to_lds_b32(
        (AS1 int*)gptr, (AS3 int*)lptr, 0, 0);
    __builtin_amdgcn_global_load_async_to_lds_b32(
        (AS1 int*)(gptr + 1), (AS3 int*)(lptr + 1), 0, 0);
#else
    lptr[0] = gptr[0];
    lptr[1] = gptr[1];
#endif
}

__device__ __forceinline__ void wait_async() {
#if defined(__AMDGCN__) && __has_builtin(__builtin_amdgcn_s_wait_asynccnt)
    __builtin_amdgcn_s_wait_asynccnt(0);
#elif defined(__AMDGCN__)
    asm volatile("s_wait_asynccnt 0" ::: "memory");
#endif
}

// ---- exact median-of-25: Devillard 99-exchange selection network ----------
#define S2(a, b)                          \
    {                                     \
        float _mn = fminf((a), (b));      \
        (b)       = fmaxf((a), (b));      \
        (a)       = _mn;                  \
    }

__device__ __forceinline__ float med25(float (&p)[25]) {
    S2(p[0],p[1]);   S2(p[3],p[4]);   S2(p[2],p[4]);
    S2(p[2],p[3]);   S2(p[6],p[7]);   S2(p[5],p[7]);
    S2(p[5],p[6]);   S2(p[9],p[10]);  S2(p[8],p[10]);
    S2(p[8],p[9]);   S2(p[12],p[13]); S2(p[11],p[13]);
    S2(p[11],p[12]); S2(p[15],p[16]); S2(p[14],p[16]);
    S2(p[14],p[15]); S2(p[18],p[19]); S2(p[17],p[19]);
    S2(p[17],p[18]); S2(p[21],p[22]); S2(p[20],p[22]);
    S2(p[20],p[21]); S2(p[23],p[24]); S2(p[2],p[5]);
    S2(p[3],p[6]);   S2(p[0],p[6]);   S2(p[0],p[3]);
    S2(p[4],p[7]);   S2(p[1],p[7]);   S2(p[1],p[4]);
    S2(p[11],p[14]); S2(p[8],p[14]);  S2(p[8],p[11]);
    S2(p[12],p[15]); S2(p[9],p[15]);  S2(p[9],p[12]);
    S2(p[13],p[16]); S2(p[10],p[16]); S2(p[10],p[13]);
    S2(p[20],p[23]); S2(p[17],p[23]); S2(p[17],p[20]);
    S2(p[21],p[24]); S2(p[18],p[24]); S2(p[18],p[21]);
    S2(p[19],p[22]); S2(p[8],p[17]);  S2(p[9],p[18]);
    S2(p[0],p[18]);  S2(p[0],p[9]);   S2(p[10],p[19]);
    S2(p[1],p[19]);  S2(p[1],p[10]);  S2(p[11],p[20]);
    S2(p[2],p[20]);  S2(p[2],p[11]);  S2(p[12],p[21]);
    S2(p[3],p[21]);  S2(p[3],p[12]);  S2(p[13],p[22]);
    S2(p[4],p[22]);  S2(p[4],p[13]);  S2(p[14],p[23]);
    S2(p[5],p[23]);  S2(p[5],p[14]);  S2(p[15],p[24]);
    S2(p[6],p[24]);  S2(p[6],p[15]);  S2(p[7],p[16]);
    S2(p[7],p[19]);  S2(p[13],p[21]); S2(p[15],p[23]);
    S2(p[7],p[13]);  S2(p[7],p[15]);  S2(p[1],p[9]);
    S2(p[3],p[11]);  S2(p[5],p[17]);  S2(p[11],p[17]);
    S2(p[9],p[17]);  S2(p[4],p[10]);  S2(p[6],p[12]);
    S2(p[7],p[14]);  S2(p[4],p[6]);   S2(p[4],p[7]);
    S2(p[12],p[14]); S2(p[10],p[14]); S2(p[6],p[7]);
    S2(p[10],p[12]); S2(p[6],p[10]);  S2(p[6],p[17]);
    S2(p[12],p[17]); S2(p[7],p[17]);  S2(p[7],p[10]);
    S2(p[12],p[18]); S2(p[7],p[12]);  S2(p[10],p[18]);
    S2(p[12],p[20]); S2(p[10],p[20]); S2(p[10],p[12]);
    return p[12];   // lower median of 25 (torch convention; exact for odd n)
}

// ---- kernel ----------------------------------------------------------------
__global__ __launch_bounds__(256) void median5x5_kernel(
        const float* __restrict__ in, float* __restrict__ out) {
    __shared__ float tile[HY][HX];

    const int    tid  = threadIdx.x;          // 0..255
    const size_t base = (size_t)blockIdx.z * (IMG * IMG);
    const float* src  = in + base;
    float*       dst  = out + base;
    const int    x0   = blockIdx.x * TX;
    const int    y0   = blockIdx.y * TY;

    const bool interior = (x0 >= 2) && (x0 + TX + 2 <= IMG) &&
                          (y0 >= 2) && (y0 + TY + 2 <= IMG);

    if (interior) {
        // Async-DMA the 68x12 halo tile into LDS as 8-byte chunks.
        // Global row start (x0-2) is even -> 8B aligned; LDS row stride 272B.
        {
            const int j   = tid;              // 0..255  (all < 408, unconditional)
            const int row = j / 34;
            const int col = (j - row * 34) * 2;
            const float* gp = src + (size_t)(y0 + row - 2) * IMG + (x0 + col - 2);
            async_copy_b64(gp, &tile[row][col]);
        }
        {
            const int j = tid + 256;          // second round: lanes with j < 408
            if (j < NCHUNK) {
                const int row = j / 34;
                const int col = (j - row * 34) * 2;
                const float* gp = src + (size_t)(y0 + row - 2) * IMG + (x0 + col - 2);
                async_copy_b64(gp, &tile[row][col]);
            }
        }
        wait_async();
    } else {
        // Boundary tile: predicated sync loads with zero padding.
        for (int i = tid; i < HX * HY; i += 256) {
            const int ly = i / HX;
            const int lx = i - ly * HX;
            const int gy = y0 + ly - 2;
            const int gx = x0 + lx - 2;
            float v = 0.0f;
            if (gy >= 0 && gy < IMG && gx >= 0 && gx < IMG)
                v = src[(size_t)gy * IMG + gx];
            tile[ly][lx] = v;
        }
    }
    __syncthreads();

    // Two ADJACENT output pixels per thread: windows share 20/25 values ->
    // only a 5x6 patch (30 LDS values, 8B-aligned pairs -> ds_load_b64),
    // while the two 99-CEX networks stay independent for dual-issue.
    const int tx2 = (tid & 31) * 2;           // local x of left pixel (even)
    const int ty  = tid >> 5;                 // 0..7

    float w[5][6];
#pragma unroll
    for (int r = 0; r < 5; ++r) {
#pragma unroll
        for (int c = 0; c < 6; ++c)
            w[r][c] = tile[ty + r][tx2 + c];
    }

    float p[25], q[25];
#pragma unroll
    for (int r = 0; r < 5; ++r) {
#pragma unroll
        for (int c = 0; c < 5; ++c) {
            p[r * 5 + c] = w[r][c];
            q[r * 5 + c] = w[r][c + 1];
        }
    }

    const float mp = med25(p);
    const float mq = med25(q);

    // One aligned 8B store per thread; output is write-once -> non-temporal.
    v2f res;
    res.x = mp;
    res.y = mq;
    v2f* outp = (v2f*)(dst + (size_t)(y0 + ty) * IMG + x0 + tx2);
    __builtin_nontemporal_store(res, outp);
}

extern "C" void kernel_launch(void* const* d_in, const int* in_sizes, int n_in,
                              void* d_out, int out_size, void* d_ws, size_t ws_size,
                              hipStream_t stream) {
    const float* x   = (const float*)d_in[0];
    float*       out = (float*)d_out;
    const int images = in_sizes[0] / (IMG * IMG);   // B*C = 48
    dim3 grid(IMG / TX, IMG / TY, images);          // 8 x 64 x 48
    median5x5_kernel<<<grid, dim3(256, 1, 1), 0, stream>>>(x, out);
}